// GCN_23252952940669
// MI455X (gfx1250) — compile-verified
//
#include <hip/hip_runtime.h>
#include <hip/hip_bf16.h>

typedef float v2f __attribute__((ext_vector_type(2)));
typedef float v4f __attribute__((ext_vector_type(4)));
typedef float v8f __attribute__((ext_vector_type(8)));

#define N_NODES 100000
#define N_EDGES 3200000
#define IN_CH   128
#define HID     16

// ---------------- degree / normalization ----------------

__global__ void k_deg_init(float* __restrict__ deg, int n) {
    int i = blockIdx.x * blockDim.x + threadIdx.x;
    if (i < n) deg[i] = 1.0f;                       // self-loop contributes 1
}

__global__ void k_deg_edges(const int* __restrict__ col, float* __restrict__ deg, int e) {
    int i = blockIdx.x * blockDim.x + threadIdx.x;
    if (i < e) atomicAdd(&deg[col[i]], 1.0f);
}

__global__ void k_dinv(float* __restrict__ deg, int n) {
    int i = blockIdx.x * blockDim.x + threadIdx.x;
    if (i < n) {
        float d = deg[i];
        deg[i] = (d > 0.0f) ? rsqrtf(d) : 0.0f;     // d >= 1 always, but keep the guard
    }
}

// ---------------- WMMA GEMM: C[n x 16] = A[n x K] @ W[K x 16] ----------------
// One wave32 computes one 16x16 output tile using V_WMMA_F32_16X16X4_F32.
// A-frag layout (16x4 f32): lanes 0-15 -> row m, K={k,k+1}; lanes 16-31 -> row m, K={k+2,k+3}.
// B-frag layout (4x16 f32): same K split, col = lane&15.
// C/D layout (16x16 f32):  vgpr i -> row (i + 8*(lane>>4)), col = lane&15.

__global__ void k_gemm_wmma(const float* __restrict__ A, const float* __restrict__ W,
                            float* __restrict__ C, int n, int K) {
    extern __shared__ float sW[];                    // K*16 floats
    int t = threadIdx.x;
    for (int i = t; i < K * HID; i += blockDim.x) sW[i] = W[i];
    __syncthreads();

    int wave = t >> 5;
    int lane = t & 31;
    int m    = lane & 15;
    int half = lane >> 4;

    int tile = blockIdx.x * (blockDim.x >> 5) + wave;
    int rb   = tile * 16;
    if (rb >= n) return;                             // wave-uniform exit (EXEC stays all-1 for WMMA)

    int row = rb + m;
    if (row > n - 1) row = n - 1;                    // N%16==0 so never taken; safety only
    const float* arow = A + (long)row * K + 2 * half;

    v8f acc = {};
    #pragma unroll 8
    for (int k = 0; k < K; k += 4) {
        v2f a = *reinterpret_cast<const v2f*>(arow + k);   // K={k+2h, k+2h+1} for this lane
        int kb = k + 2 * half;
        v2f b;
        b[0] = sW[kb * HID + m];
        b[1] = sW[(kb + 1) * HID + m];
        acc = __builtin_amdgcn_wmma_f32_16x16x4_f32(
                  /*neg_a=*/false, a, /*neg_b=*/false, b,
                  /*c_mod=*/(short)0, acc, /*reuse_a=*/false, /*reuse_b=*/false);
    }

    #pragma unroll
    for (int i = 0; i < 8; ++i) {
        int r = rb + i + 8 * half;
        if (r < n) C[(long)r * HID + m] = acc[i];
    }
}

// ---------------- aggregation ----------------

__global__ void k_zero4(v4f* __restrict__ buf, int n4) {
    int i = blockIdx.x * blockDim.x + threadIdx.x;
    if (i < n4) buf[i] = (v4f){0.f, 0.f, 0.f, 0.f};
}

// 4 lanes per edge, 4 channels each: float4 gather + 4 float atomics scatter.
__global__ void k_edge_agg(const int* __restrict__ row, const int* __restrict__ col,
                           const float* __restrict__ dinv, const float* __restrict__ h,
                           float* __restrict__ agg, int e) {
    int t = blockIdx.x * blockDim.x + threadIdx.x;
    if (t >= 4 * e) return;
    int eid = t >> 2;
    int c4  = (t & 3) << 2;
    int r = row[eid];
    int c = col[eid];
    float w = dinv[r] * dinv[c];
    v4f hv = *reinterpret_cast<const v4f*>(h + (long)r * HID + c4);
    float* dst = agg + (long)c * HID + c4;
    atomicAdd(dst + 0, w * hv[0]);
    atomicAdd(dst + 1, w * hv[1]);
    atomicAdd(dst + 2, w * hv[2]);
    atomicAdd(dst + 3, w * hv[3]);
}

// agg[i,c] = relu(agg[i,c] + dinv[i]^2 * h[i,c] + b[c])   (self-loop + bias + relu, in place)
__global__ void k_self_bias_relu(float* __restrict__ agg, const float* __restrict__ h,
                                 const float* __restrict__ dinv, const float* __restrict__ b,
                                 int n) {
    int t = blockIdx.x * blockDim.x + threadIdx.x;
    if (t >= n * HID) return;
    int i = t >> 4;
    int c = t & 15;
    float di = dinv[i];
    float v = agg[t] + di * di * h[t] + b[c];
    agg[t] = v > 0.0f ? v : 0.0f;
}

// out[i] = dot(h[i,:], Wl) + bl
__global__ void k_linear(const float* __restrict__ h, const float* __restrict__ Wl,
                         const float* __restrict__ bl, float* __restrict__ out, int n) {
    int i = blockIdx.x * blockDim.x + threadIdx.x;
    if (i >= n) return;
    const float* hr = h + (long)i * HID;
    float s = bl[0];
    #pragma unroll
    for (int c = 0; c < HID; ++c) s += hr[c] * Wl[c];
    out[i] = s;
}

// ---------------- launch ----------------

extern "C" void kernel_launch(void* const* d_in, const int* in_sizes, int n_in,
                              void* d_out, int out_size, void* d_ws, size_t ws_size,
                              hipStream_t stream) {
    const float* x  = (const float*)d_in[0];
    const int*   ei = (const int*)d_in[1];
    const int*   row = ei;                 // edge_index[0] : sources
    const int*   col = ei + N_EDGES;       // edge_index[1] : targets
    const float* W1 = (const float*)d_in[2];
    const float* b1 = (const float*)d_in[3];
    const float* W2 = (const float*)d_in[4];
    const float* b2 = (const float*)d_in[5];
    const float* Wl = (const float*)d_in[6];
    const float* bl = (const float*)d_in[7];
    float* out = (float*)d_out;

    float* ws   = (float*)d_ws;
    float* dinv = ws;                                          // N floats
    float* bufA = ws + ((N_NODES + 255) / 256) * 256;          // N*16 floats
    float* bufB = bufA + (size_t)N_NODES * HID;                // N*16 floats

    const int T = 256;
    const int gN   = (N_NODES + T - 1) / T;
    const int gE   = (N_EDGES + T - 1) / T;
    const int gE4  = (4 * N_EDGES + T - 1) / T;
    const int gNH  = (N_NODES * HID + T - 1) / T;
    const int gNH4 = (N_NODES * HID / 4 + T - 1) / T;
    const int tiles = (N_NODES + 15) / 16;
    const int gGemm = (tiles + (T / 32) - 1) / (T / 32);

    // degrees (A + I, counted at target) -> dinv
    k_deg_init<<<gN, T, 0, stream>>>(dinv, N_NODES);
    k_deg_edges<<<gE, T, 0, stream>>>(col, dinv, N_EDGES);
    k_dinv<<<gN, T, 0, stream>>>(dinv, N_NODES);

    // ---- layer 1 ----
    k_gemm_wmma<<<gGemm, T, IN_CH * HID * sizeof(float), stream>>>(x, W1, bufA, N_NODES, IN_CH);
    k_zero4<<<gNH4, T, 0, stream>>>((v4f*)bufB, N_NODES * HID / 4);
    k_edge_agg<<<gE4, T, 0, stream>>>(row, col, dinv, bufA, bufB, N_EDGES);
    k_self_bias_relu<<<gNH, T, 0, stream>>>(bufB, bufA, dinv, b1, N_NODES);   // bufB = h1

    // ---- layer 2 ----
    k_gemm_wmma<<<gGemm, T, HID * HID * sizeof(float), stream>>>(bufB, W2, bufA, N_NODES, HID);
    k_zero4<<<gNH4, T, 0, stream>>>((v4f*)bufB, N_NODES * HID / 4);
    k_edge_agg<<<gE4, T, 0, stream>>>(row, col, dinv, bufA, bufB, N_EDGES);
    k_self_bias_relu<<<gNH, T, 0, stream>>>(bufB, bufA, dinv, b2, N_NODES);   // bufB = h2

    // ---- readout ----
    k_linear<<<gN, T, 0, stream>>>(bufB, Wl, bl, out, N_NODES);
}